// FixedRevRNN_63720134803724
// MI455X (gfx1250) — compile-verified
//
#include <hip/hip_runtime.h>

// Problem constants (from reference): B=8, T=256, F=128, V=256, DEPTH=1, OUT=256, DELAY=8
#define RR_B     8
#define RR_T     256
#define RR_F     128
#define RR_V     256
#define RR_DELAY 8
#define RR_OUT   256
#define RR_NSTEP (RR_T + RR_DELAY)
#define RR_MAT   (RR_F * RR_F)      // 16384 floats = 64 KB

typedef float v2f __attribute__((ext_vector_type(2)));
typedef float v8f __attribute__((ext_vector_type(8)));

__device__ __forceinline__ v8f wmma_f32(v2f a, v2f b, v8f c) {
    // V_WMMA_F32_16X16X4_F32 : D(16x16,f32) = A(16x4,f32) x B(4x16,f32) + C
    return __builtin_amdgcn_wmma_f32_16x16x4_f32(false, a, false, b, (short)0, c,
                                                 false, false);
}

// LDS bank swizzle: element (r,c) of a 128x128 LDS matrix lives at
//   r*128 + (c ^ ((r & 15) << 2))
// XOR touches only bits >= 2 of the column, so (c, c+1) float2 pairs stay
// contiguous and 8B-aligned. This makes the WMMA A-fragment reads (16 rows at
// one column) conflict-free across the 64 LDS banks instead of 16-way
// conflicted, and makes the C-fragment writes conflict-free as well.
__device__ __forceinline__ int lds_swz_i(int i) {           // linear index form
    return i ^ ((((unsigned)i >> 7) & 15) << 2);
}

// ---------------------------------------------------------------------------
// C = A @ B for 128x128x128 fp32, row-major. One 256-thread block (8 waves).
// Wave w computes output rows [16w, 16w+16); loops over 8 column tiles.
// Template flags select swizzled (LDS) vs linear (global) layout per operand.
// ---------------------------------------------------------------------------
template <bool ASWZ, bool BSWZ, bool CSWZ>
__device__ __forceinline__ void mm128(const float* __restrict__ A,
                                      const float* __restrict__ B,
                                      float* __restrict__ C) {
    const int wave = threadIdx.x >> 5;
    const int lane = threadIdx.x & 31;
    const int half = lane >> 4;      // 0: lanes 0-15, 1: lanes 16-31
    const int l16  = lane & 15;
    const int mrow = wave * 16 + l16;
    const int asw  = ASWZ ? ((mrow & 15) << 2) : 0;
    const float* arow = A + mrow * RR_F;
#pragma unroll 1
    for (int tn = 0; tn < 8; ++tn) {
        const int ncol = tn * 16 + l16;
        v8f acc = {0.f, 0.f, 0.f, 0.f, 0.f, 0.f, 0.f, 0.f};
#pragma unroll 8
        for (int kb = 0; kb < RR_F; kb += 4) {
            const int k0 = kb + half * 2;      // K index of VGPR0 for this lane
            const int ka = k0 ^ asw;           // swizzled column (pair-preserving)
            v2f a;
            a.x = arow[ka];
            a.y = arow[ka + 1];
            const int bc0 = BSWZ ? (ncol ^ ((k0 & 15) << 2)) : ncol;
            const int bc1 = BSWZ ? (ncol ^ (((k0 + 1) & 15) << 2)) : ncol;
            v2f b;
            b.x = B[k0 * RR_F + bc0];
            b.y = B[(k0 + 1) * RR_F + bc1];
            acc = wmma_f32(a, b, acc);
        }
        const int rbase = wave * 16 + half * 8;
#pragma unroll
        for (int r = 0; r < 8; ++r) {
            const int rr = rbase + r;
            const int cc = CSWZ ? (ncol ^ ((rr & 15) << 2)) : ncol;
            C[rr * RR_F + cc] = acc[r];
        }
    }
}

// ---------------------------------------------------------------------------
// Precompute D[which][v] = (emb[v] @ lp_which[128:,:]) @ activate   (v==V -> I)
// Hoists all token-dependent work out of the sequential scan.
// Grid: (V+1, 2), block 256.
// ---------------------------------------------------------------------------
__global__ void __launch_bounds__(256)
precompute_D(const float* __restrict__ emb, const float* __restrict__ lp0,
             const float* __restrict__ lp1, const float* __restrict__ act,
             float* __restrict__ Dws) {
    __shared__ float E[RR_MAT];
    __shared__ float Tm[RR_MAT];
    const int v     = blockIdx.x;   // 0..V   (V = identity, used by delay steps)
    const int which = blockIdx.y;   // 0 -> lp0, 1 -> lp1
    const float* lph = (which ? lp1 : lp0) + RR_MAT;   // high half [128:,:]
    float* D = Dws + ((size_t)which * (RR_V + 1) + v) * RR_MAT;

    if (v < RR_V) {
        const float* ev = emb + (size_t)v * RR_MAT;
        for (int i = threadIdx.x; i < RR_MAT; i += 256) E[lds_swz_i(i)] = ev[i];
    } else {
        for (int i = threadIdx.x; i < RR_MAT; i += 256)
            E[lds_swz_i(i)] = ((i >> 7) == (i & 127)) ? 1.0f : 0.0f;
    }
    __syncthreads();
    mm128<true, false, true>(E, lph, Tm);    // emb[v] @ lp_hi       -> LDS
    __syncthreads();
    mm128<true, false, false>(Tm, act, D);   // (..) @ activate      -> global
}

// ---------------------------------------------------------------------------
// The sequential reversible scan. One workgroup per batch element.
// Five 64KB fp32 matrices fill the WGP's 320KB LDS exactly (bank-swizzled):
//   s0,s1 = state, t1,t2,t3 = temporaries (rotating).
// Per step (6 chained 128^3 WMMA matmuls):
//   t1 = s0@A0 ; t2 = t1@D0[tok] ; t3 = s1@t2 (out1)
//   t1 = t3@A1 ; t2 = t1@D1[tok] ; s1 = s0@t2 (out0)  ; rotate (s0,s1,t3)<-(s1,t3,s0)
// Next step's 128KB of D tables is prefetched into cache while computing.
// Steps >= DELAY emit ys[t= s-DELAY, b] = [out0 ; out1] to global workspace.
// ---------------------------------------------------------------------------
__global__ void __launch_bounds__(256)
rev_scan(const int* __restrict__ tok, const float* __restrict__ lp0,
         const float* __restrict__ lp1, const float* __restrict__ hidden,
         const float* __restrict__ Dws, float* __restrict__ Y) {
    __shared__ float P[5][RR_MAT];   // 5 * 64 KB = 320 KB (full WGP LDS)
    const int b = blockIdx.x;
    const float* D0 = Dws;
    const float* D1 = Dws + (size_t)(RR_V + 1) * RR_MAT;

    // init state: broadcast hidden[:128] / hidden[128:]
    for (int i = threadIdx.x; i < RR_MAT; i += 256) {
        const int si = lds_swz_i(i);
        P[0][si] = hidden[i];
        P[1][si] = hidden[RR_MAT + i];
    }
    __syncthreads();

    int s0 = 0, s1 = 1, t3 = 4;
    const int t1 = 2, t2 = 3;
    for (int s = 0; s < RR_NSTEP; ++s) {
        const int v = (s < RR_T) ? tok[b * RR_T + s] : RR_V;
        const float* d0 = D0 + (size_t)v * RR_MAT;
        const float* d1 = D1 + (size_t)v * RR_MAT;

        // Prefetch next step's D tables (256 B per thread covers 2x64KB).
        if (s + 1 < RR_NSTEP) {
            const int vn = (s + 1 < RR_T) ? tok[b * RR_T + s + 1] : RR_V;
            const float* p0 = D0 + (size_t)vn * RR_MAT + threadIdx.x * 64;
            const float* p1 = D1 + (size_t)vn * RR_MAT + threadIdx.x * 64;
            __builtin_prefetch(p0, 0, 1);
            __builtin_prefetch(p0 + 32, 0, 1);
            __builtin_prefetch(p1, 0, 1);
            __builtin_prefetch(p1 + 32, 0, 1);
        }

        mm128<true, false, true>(P[s0], lp0, P[t1]); __syncthreads(); // s0 @ A0
        mm128<true, false, true>(P[t1], d0,  P[t2]); __syncthreads(); // .. @ D0[tok]
        mm128<true, true,  true>(P[s1], P[t2], P[t3]); __syncthreads(); // out1
        mm128<true, false, true>(P[t3], lp1, P[t1]); __syncthreads(); // out1 @ A1
        mm128<true, false, true>(P[t1], d1,  P[t2]); __syncthreads(); // .. @ D1[tok]
        mm128<true, true,  true>(P[s0], P[t2], P[s1]); __syncthreads(); // out0

        const int ns0 = s1, ns1 = t3, nt3 = s0;        // rotate buffers
        s0 = ns0; s1 = ns1; t3 = nt3;

        if (s >= RR_DELAY) {
            float* yrow = Y + ((size_t)(s - RR_DELAY) * RR_B + b) * (2 * RR_MAT);
            for (int i = threadIdx.x; i < RR_MAT; i += 256) {
                const int si = lds_swz_i(i);
                yrow[i]          = P[s0][si];   // out0 rows 0..127
                yrow[RR_MAT + i] = P[s1][si];   // out1 rows 128..255
            }
            __syncthreads();
        }
    }
}

// ---------------------------------------------------------------------------
// Output projection: out[2048 x 256] = Y[2048 x 32768] @ W[32768 x 256].
// (The reference's reshape makes row r = t*8+b land contiguously, so this is a
//  plain GEMM.) Grid (16,2,8): 8-way K-split, fp32 atomic accumulation.
// ---------------------------------------------------------------------------
__global__ void __launch_bounds__(256)
proj_gemm(const float* __restrict__ Y, const float* __restrict__ W,
          float* __restrict__ out) {
    const int mt = blockIdx.x;          // row block of 128 (16 total)
    const int nt = blockIdx.y;          // col block of 128 (2 total)
    const int kt = blockIdx.z;          // K segment (8 total, 4096 each)
    const int wave = threadIdx.x >> 5;
    const int lane = threadIdx.x & 31;
    const int half = lane >> 4;
    const int l16  = lane & 15;
    const int K = 2 * RR_MAT;           // 32768
    const float* arow = Y + (size_t)(mt * 128 + wave * 16 + l16) * K;
    const int kbeg = kt * (K / 8);
#pragma unroll 1
    for (int tn = 0; tn < 8; ++tn) {
        const int col = nt * 128 + tn * 16 + l16;
        v8f acc = {0.f, 0.f, 0.f, 0.f, 0.f, 0.f, 0.f, 0.f};
#pragma unroll 4
        for (int kb = kbeg; kb < kbeg + K / 8; kb += 4) {
            const int k0 = kb + half * 2;
            v2f a;
            a.x = arow[k0];
            a.y = arow[k0 + 1];
            v2f bb;
            bb.x = W[(size_t)k0 * RR_OUT + col];
            bb.y = W[(size_t)(k0 + 1) * RR_OUT + col];
            acc = wmma_f32(a, bb, acc);
        }
        const int rbase = mt * 128 + wave * 16 + half * 8;
#pragma unroll
        for (int r = 0; r < 8; ++r)
            atomicAdd(&out[(size_t)(rbase + r) * RR_OUT + col], acc[r]);
    }
}

__global__ void zero_f32(float* __restrict__ p, int n) {
    int i = blockIdx.x * blockDim.x + threadIdx.x;
    if (i < n) p[i] = 0.0f;
}

// ---------------------------------------------------------------------------
// inputs (setup_inputs order):
//   0 fn_input  int32  [B,T]
//   1 embedding f32    [V,F,F]
//   2 lp0       f32    [1,2F,F]
//   3 lp1       f32    [1,2F,F]
//   4 hidden    f32    [2F,F]
//   5 activate  f32    [F,F]
//   6 out_linear f32   [2F*F, OUT]
// output: f32 [B,T,OUT] = 524288 floats
// workspace: D tables 2*(V+1)*MAT floats (~33.7MB) + Y (T*B*2*MAT floats, 256MB)
// ---------------------------------------------------------------------------
extern "C" void kernel_launch(void* const* d_in, const int* in_sizes, int n_in,
                              void* d_out, int out_size, void* d_ws, size_t ws_size,
                              hipStream_t stream) {
    const int*   tok    = (const int*)d_in[0];
    const float* emb    = (const float*)d_in[1];
    const float* lp0    = (const float*)d_in[2];
    const float* lp1    = (const float*)d_in[3];
    const float* hidden = (const float*)d_in[4];
    const float* act    = (const float*)d_in[5];
    const float* W      = (const float*)d_in[6];
    float* out = (float*)d_out;

    float* Dws = (float*)d_ws;                               // 2*(V+1)*MAT floats
    float* Y   = Dws + (size_t)2 * (RR_V + 1) * RR_MAT;      // T*B*2*MAT floats

    zero_f32<<<(out_size + 255) / 256, 256, 0, stream>>>(out, out_size);
    precompute_D<<<dim3(RR_V + 1, 2), 256, 0, stream>>>(emb, lp0, lp1, act, Dws);
    rev_scan<<<dim3(RR_B), 256, 0, stream>>>(tok, lp0, lp1, hidden, Dws, Y);
    proj_gemm<<<dim3(16, 2, 8), 256, 0, stream>>>(Y, W, out);
}